// LightGCNHetero_61632780698018
// MI455X (gfx1250) — compile-verified
//
#include <hip/hip_runtime.h>
#include <hip/hip_bf16.h>

// ---------------------------------------------------------------------------
// LightGCNHetero propagate for MI455X (gfx1250, wave32).
//
// 3 layers of COO SpMM (4M edges, d=64) + layer-mean pooling.
// Memory/atomic bound: ~2 GB of gather/scatter per layer, all of which can
// resolve in the 192 MB L2 (one embedding table copy = 140.8 MB).
//   - 16 lanes per edge, float4 per lane -> 256B coalesced gather/scatter
//   - scatter-add via native f32 atomics (global_atomic_add_f32, RMW in L2)
//   - layer 0 gathers straight from the three input tables (no concat copy)
//   - x1,x2,x3 kept in separate ws buffers -> pooling fused into one epilogue
// Epilogue out = (x0+x1+x2+x3) * 0.25 with the scale done as a GEMM against
// 0.25*I64 using V_WMMA_F32_16X16X4_F32 (exact: power-of-two scale, zero
// products exact; sum order matches the reference).
// ---------------------------------------------------------------------------

#define EMBED_DIM 64

typedef __attribute__((ext_vector_type(2))) float v2f;
typedef __attribute__((ext_vector_type(8))) float v8f;

// ---------------- SpMM layer 0: gather from the 3 raw tables --------------
__global__ __launch_bounds__(256) void lightgcn_spmm_l0_kernel(
    const float* __restrict__ user_w, const float* __restrict__ item_w,
    const float* __restrict__ author_w, int n_users, int n_items,
    float* __restrict__ y, const float* __restrict__ vals,
    const int* __restrict__ rows, const int* __restrict__ cols, int n_edges) {
  unsigned tid = blockIdx.x * blockDim.x + threadIdx.x;
  int e = (int)(tid >> 4);
  if (e >= n_edges) return;
  int sub = (int)(tid & 15);

  int pf = e + 4096;
  if (pf < n_edges) {              // stream-prefetch edge arrays (gfx1250)
    __builtin_prefetch(&rows[pf], 0, 1);
    __builtin_prefetch(&cols[pf], 0, 1);
    __builtin_prefetch(&vals[pf], 0, 1);
  }

  int   c = cols[e];
  int   r = rows[e];
  float v = vals[e];

  // concat-space column -> source table (uniform across the 16-lane group)
  const float* src;
  if (c < n_users)                src = user_w   + (size_t)c * EMBED_DIM;
  else if (c < n_users + n_items) src = item_w   + (size_t)(c - n_users) * EMBED_DIM;
  else                            src = author_w + (size_t)(c - n_users - n_items) * EMBED_DIM;

  float4 xv = ((const float4*)src)[sub];
  float* yr = y + (size_t)r * EMBED_DIM + (size_t)sub * 4;
  unsafeAtomicAdd(yr + 0, v * xv.x);
  unsafeAtomicAdd(yr + 1, v * xv.y);
  unsafeAtomicAdd(yr + 2, v * xv.z);
  unsafeAtomicAdd(yr + 3, v * xv.w);
}

// ---------------- SpMM: y[rows[e]] += vals[e] * x[cols[e]] ----------------
__global__ __launch_bounds__(256) void lightgcn_spmm_kernel(
    const float* __restrict__ x, float* __restrict__ y,
    const float* __restrict__ vals, const int* __restrict__ rows,
    const int* __restrict__ cols, int n_edges) {
  unsigned tid = blockIdx.x * blockDim.x + threadIdx.x;
  int e = (int)(tid >> 4);
  if (e >= n_edges) return;
  int sub = (int)(tid & 15);

  int pf = e + 4096;
  if (pf < n_edges) {
    __builtin_prefetch(&rows[pf], 0, 1);
    __builtin_prefetch(&cols[pf], 0, 1);
    __builtin_prefetch(&vals[pf], 0, 1);
  }

  int   c = cols[e];
  int   r = rows[e];
  float v = vals[e];

  float4 xv = ((const float4*)(x + (size_t)c * EMBED_DIM))[sub];
  float* yr = y + (size_t)r * EMBED_DIM + (size_t)sub * 4;
  unsafeAtomicAdd(yr + 0, v * xv.x);
  unsafeAtomicAdd(yr + 1, v * xv.y);
  unsafeAtomicAdd(yr + 2, v * xv.z);
  unsafeAtomicAdd(yr + 3, v * xv.w);
}

// ------- fused pooling epilogue: out = (x0+x1+x2+x3) x (0.25*I64) ---------
// Each wave owns one 16x64 row block. Per 16x16 tile: D = A x B + C over the
// 4 diagonal K-chunks, B = 4x16 slices of 0.25*I.
// Layouts per CDNA5 ISA 7.12.2 (wave32):
//   A 16x4:  lanes 0-15 M=lane  {VGPR0=K0, VGPR1=K1};
//            lanes16-31 M=lane-16 {VGPR0=K2, VGPR1=K3}
//   C/D 16x16: VGPR d holds M = d + 8*(lane>=16), N = lane&15
__global__ __launch_bounds__(256) void lightgcn_pool_wmma_kernel(
    const float* __restrict__ user_w, const float* __restrict__ item_w,
    const float* __restrict__ x1, const float* __restrict__ x2,
    const float* __restrict__ x3, float* __restrict__ out,
    int n_users, int n_rowblocks) {
  int wave = blockIdx.x * (blockDim.x >> 5) + (threadIdx.x >> 5);
  if (wave >= n_rowblocks) return;   // uniform per wave -> EXEC stays all-1s
  int lane  = threadIdx.x & 31;
  int m     = lane & 15;
  int khalf = (lane >> 4) << 1;      // 0 for lanes 0-15, 2 for lanes 16-31

  size_t row = (size_t)wave * 16 + m;
  const float* r0 = (row < (size_t)n_users)
                        ? (user_w + row * EMBED_DIM)
                        : (item_w + (row - (size_t)n_users) * EMBED_DIM);
  const float* r1 = x1 + row * EMBED_DIM;
  const float* r2 = x2 + row * EMBED_DIM;
  const float* r3 = x3 + row * EMBED_DIM;
  float* obase = out + (size_t)wave * 16 * EMBED_DIM;

#pragma unroll
  for (int j = 0; j < 4; ++j) {
    v8f c = {};
#pragma unroll
    for (int t = 0; t < 4; ++t) {
      int k = 16 * j + 4 * t + khalf;
      v2f a;                          // reference order: ((x0+x1)+x2)+x3
      a.x = ((r0[k]     + r1[k])     + r2[k])     + r3[k];
      a.y = ((r0[k + 1] + r1[k + 1]) + r2[k + 1]) + r3[k + 1];
      v2f b;                          // B[r][n] = 0.25 iff n == 4t + r
      b.x = (m == 4 * t + khalf)     ? 0.25f : 0.0f;
      b.y = (m == 4 * t + khalf + 1) ? 0.25f : 0.0f;
      c = __builtin_amdgcn_wmma_f32_16x16x4_f32(
          false, a, false, b, (short)0, c, false, false);
    }
    int rowoff = (lane >> 4) * 8;
#pragma unroll
    for (int d = 0; d < 8; ++d)
      obase[(size_t)(rowoff + d) * EMBED_DIM + 16 * j + m] = c[d];
  }
}

// scalar tail for row counts not divisible by 16 (not hit for this problem)
__global__ __launch_bounds__(256) void lightgcn_pool_tail_kernel(
    const float* __restrict__ user_w, const float* __restrict__ item_w,
    const float* __restrict__ x1, const float* __restrict__ x2,
    const float* __restrict__ x3, float* __restrict__ out,
    int n_users, int start_row, int n_rows) {
  int i = blockIdx.x * blockDim.x + threadIdx.x;
  int total = (n_rows - start_row) * EMBED_DIM;
  if (i >= total) return;
  size_t row = (size_t)start_row + (size_t)(i / EMBED_DIM);
  int    d   = i % EMBED_DIM;
  float x0 = (row < (size_t)n_users)
                 ? user_w[row * EMBED_DIM + d]
                 : item_w[(row - (size_t)n_users) * EMBED_DIM + d];
  size_t idx = row * EMBED_DIM + d;
  out[idx] = (((x0 + x1[idx]) + x2[idx]) + x3[idx]) * 0.25f;
}

// ---------------- fallback-path kernels (small workspace) -----------------
__global__ __launch_bounds__(256) void lightgcn_acc_kernel(
    float* __restrict__ acc, const float* __restrict__ y, int n4) {
  int i = blockIdx.x * blockDim.x + threadIdx.x;
  if (i >= n4) return;
  float4 a = ((const float4*)acc)[i];
  float4 b = ((const float4*)y)[i];
  a.x += b.x; a.y += b.y; a.z += b.z; a.w += b.w;
  ((float4*)acc)[i] = a;
}

__global__ __launch_bounds__(256) void lightgcn_scale_wmma_kernel(
    float* __restrict__ out, int n_rowblocks) {
  int wave = blockIdx.x * (blockDim.x >> 5) + (threadIdx.x >> 5);
  if (wave >= n_rowblocks) return;
  int lane  = threadIdx.x & 31;
  int m     = lane & 15;
  int khalf = (lane >> 4) << 1;
  float* base = out + (size_t)wave * 16 * EMBED_DIM;
#pragma unroll
  for (int j = 0; j < 4; ++j) {
    v8f c = {};
#pragma unroll
    for (int t = 0; t < 4; ++t) {
      int k = 16 * j + 4 * t + khalf;
      v2f a;
      a.x = base[(size_t)m * EMBED_DIM + k];
      a.y = base[(size_t)m * EMBED_DIM + k + 1];
      v2f b;
      b.x = (m == 4 * t + khalf)     ? 0.25f : 0.0f;
      b.y = (m == 4 * t + khalf + 1) ? 0.25f : 0.0f;
      c = __builtin_amdgcn_wmma_f32_16x16x4_f32(
          false, a, false, b, (short)0, c, false, false);
    }
    int rowoff = (lane >> 4) * 8;
#pragma unroll
    for (int d = 0; d < 8; ++d)
      base[(size_t)(rowoff + d) * EMBED_DIM + 16 * j + m] = c[d];
  }
}

__global__ __launch_bounds__(256) void lightgcn_scale_tail_kernel(
    float* __restrict__ out, int start, int n) {
  int i = start + blockIdx.x * blockDim.x + threadIdx.x;
  if (i < n) out[i] *= 0.25f;
}

extern "C" void kernel_launch(void* const* d_in, const int* in_sizes, int n_in,
                              void* d_out, int out_size, void* d_ws, size_t ws_size,
                              hipStream_t stream) {
  const float* user_w   = (const float*)d_in[0];
  const float* item_w   = (const float*)d_in[1];
  const float* author_w = (const float*)d_in[2];
  const float* vals     = (const float*)d_in[3];
  const int*   rows     = (const int*)d_in[4];
  const int*   cols     = (const int*)d_in[5];

  const int n_users   = in_sizes[0] / EMBED_DIM;
  const int n_items   = in_sizes[1] / EMBED_DIM;
  const int n_authors = in_sizes[2] / EMBED_DIM;
  const int n_edges   = in_sizes[3];
  const int n_nodes   = n_users + n_items + n_authors;
  const int n_outrows = n_users + n_items;

  const size_t node_elems = (size_t)n_nodes * EMBED_DIM;
  const size_t node_bytes = node_elems * sizeof(float);
  const size_t out_elems  = (size_t)n_outrows * EMBED_DIM;

  float* out = (float*)d_out;

  const int spmm_blocks = (n_edges * 16 + 255) / 256;
  const int n_rowblocks = n_outrows / 16;
  const int wmma_blocks = (n_rowblocks + 7) / 8;     // 8 waves per block
  const int tail_rows   = n_outrows - n_rowblocks * 16;

  if (ws_size >= 3 * node_bytes) {
    // ---- fast path: x1,x2,x3 kept live; pooling fused into epilogue ----
    float* X1 = (float*)d_ws;
    float* X2 = X1 + node_elems;
    float* X3 = X2 + node_elems;

    hipMemsetAsync(X1, 0, node_bytes, stream);
    lightgcn_spmm_l0_kernel<<<spmm_blocks, 256, 0, stream>>>(
        user_w, item_w, author_w, n_users, n_items, X1, vals, rows, cols,
        n_edges);

    hipMemsetAsync(X2, 0, node_bytes, stream);
    lightgcn_spmm_kernel<<<spmm_blocks, 256, 0, stream>>>(X1, X2, vals, rows,
                                                          cols, n_edges);

    hipMemsetAsync(X3, 0, node_bytes, stream);
    lightgcn_spmm_kernel<<<spmm_blocks, 256, 0, stream>>>(X2, X3, vals, rows,
                                                          cols, n_edges);

    if (n_rowblocks > 0)
      lightgcn_pool_wmma_kernel<<<wmma_blocks, 256, 0, stream>>>(
          user_w, item_w, X1, X2, X3, out, n_users, n_rowblocks);
    if (tail_rows > 0) {
      int tn = tail_rows * EMBED_DIM;
      lightgcn_pool_tail_kernel<<<(tn + 255) / 256, 256, 0, stream>>>(
          user_w, item_w, X1, X2, X3, out, n_users, n_rowblocks * 16,
          n_outrows);
    }
  } else {
    // ---- fallback: ping-pong buffers + accumulate into d_out ----
    float* X = (float*)d_ws;
    float* Y = X + node_elems;

    hipMemcpyAsync(X, user_w, (size_t)in_sizes[0] * sizeof(float),
                   hipMemcpyDeviceToDevice, stream);
    hipMemcpyAsync(X + (size_t)n_users * EMBED_DIM, item_w,
                   (size_t)in_sizes[1] * sizeof(float),
                   hipMemcpyDeviceToDevice, stream);
    hipMemcpyAsync(X + (size_t)(n_users + n_items) * EMBED_DIM, author_w,
                   (size_t)in_sizes[2] * sizeof(float),
                   hipMemcpyDeviceToDevice, stream);
    hipMemcpyAsync(out, X, out_elems * sizeof(float),
                   hipMemcpyDeviceToDevice, stream);

    const int acc_n4     = (int)(out_elems / 4);
    const int acc_blocks = (acc_n4 + 255) / 256;
    for (int layer = 0; layer < 3; ++layer) {
      hipMemsetAsync(Y, 0, node_bytes, stream);
      lightgcn_spmm_kernel<<<spmm_blocks, 256, 0, stream>>>(X, Y, vals, rows,
                                                            cols, n_edges);
      lightgcn_acc_kernel<<<acc_blocks, 256, 0, stream>>>(out, Y, acc_n4);
      float* tmp = X; X = Y; Y = tmp;
    }
    if (n_rowblocks > 0)
      lightgcn_scale_wmma_kernel<<<wmma_blocks, 256, 0, stream>>>(out,
                                                                  n_rowblocks);
    if (tail_rows > 0) {
      int tn = tail_rows * EMBED_DIM;
      lightgcn_scale_tail_kernel<<<(tn + 255) / 256, 256, 0, stream>>>(
          out, n_rowblocks * 16 * EMBED_DIM, (int)out_elems);
    }
  }
}